// SVDEqualizer_57878979281599
// MI455X (gfx1250) — compile-verified
//
#include <hip/hip_runtime.h>

typedef __attribute__((ext_vector_type(2))) float v2f;
typedef __attribute__((ext_vector_type(8))) float v8f;

#define B_   16
#define NRA_ 4
#define NTA_ 4
#define S_   14
#define F_   4096
#define NSWEEP 4

// LDS strides padded to kill bank conflicts in the WMMA operand-gather phase
#define WSTRIDE 34     // per-site W storage stride (floats); 32 used
#define YSTRIDE 257    // per-(a,ri) y stream stride (floats); 256 used

__launch_bounds__(256)
__global__ void svd_eq_jacobi_wmma_kernel(const float* __restrict__ yr_g,
                                          const float* __restrict__ yi_g,
                                          const float* __restrict__ hr_g,
                                          const float* __restrict__ hi_g,
                                          float2* __restrict__ out2)
{
    __shared__ float lds_y[8 * YSTRIDE];      //  8.2 KB : [(a*2+ri)][f']
    __shared__ float lds_w[256 * WSTRIDE];    // 34.8 KB : per site [(ri*4+m)*4+k]
    __shared__ float lds_lam[256 * 4];        //  4.0 KB : per site lambda_k = s_k^2

    const int t     = threadIdx.x;
    const int blk   = blockIdx.x;
    const int ftile = blk & 15;               // F/256 = 16 tiles
    const int s     = (blk >> 4) % S_;
    const int b     = blk / (16 * S_);
    const int f0    = ftile * 256;
    const int f     = f0 + t;

    // ---- stage y into LDS (coalesced: 8 streams of 256 floats) ----
    #pragma unroll
    for (int a = 0; a < NRA_; ++a) {
        int yidx = ((b * NRA_ + a) * S_ + s) * F_ + f;
        lds_y[(a * 2 + 0) * YSTRIDE + t] = yr_g[yidx];
        lds_y[(a * 2 + 1) * YSTRIDE + t] = yi_g[yidx];
    }

    // ---- load this site's 4x4 complex H into registers (coalesced along f) ----
    float wr[4][4], wi[4][4];                 // [m][k], becomes W = U*S after Jacobi
    #pragma unroll
    for (int m = 0; m < 4; ++m)
        #pragma unroll
        for (int k = 0; k < 4; ++k) {
            int hidx = (((b * 4 + m) * 4 + k) * S_ + s) * F_ + f;
            wr[m][k] = hr_g[hidx];
            wi[m][k] = hi_g[hidx];
        }

    // ---- one-sided complex Jacobi: orthogonalize columns of H ----
    // After convergence, column k = s_k * u_k, so u_k^H y / s_k = w_k^H y / lambda_k.
    #pragma unroll
    for (int sw = 0; sw < NSWEEP; ++sw) {
        #pragma unroll
        for (int p = 0; p < 3; ++p) {
            #pragma unroll
            for (int q = p + 1; q < 4; ++q) {
                float cr = 0.f, ci = 0.f, aa = 0.f, bb = 0.f;
                #pragma unroll
                for (int m = 0; m < 4; ++m) {
                    float pr = wr[m][p], pi = wi[m][p];
                    float qr = wr[m][q], qi = wi[m][q];
                    cr += pr * qr + pi * qi;      // Re(w_p^H w_q)
                    ci += pr * qi - pi * qr;      // Im(w_p^H w_q)
                    aa += pr * pr + pi * pi;
                    bb += qr * qr + qi * qi;
                }
                float c2  = cr * cr + ci * ci;
                bool  rot = c2 > 1e-28f * aa * bb;        // branchless skip
                float cab = __builtin_amdgcn_sqrtf(c2);
                float inv = __builtin_amdgcn_rcpf(cab);
                float er  = cr * inv, ei = ci * inv;      // e^{i phi}
                float tau = (aa - bb) * 0.5f * inv;       // tan(2th) = 2|c|/(a-b)
                float rt  = __builtin_amdgcn_sqrtf(tau * tau + 1.0f);
                float tt  = (tau >= 0.f ? 1.0f : -1.0f) *
                            __builtin_amdgcn_rcpf(fabsf(tau) + rt);
                float cs  = __builtin_amdgcn_rsqf(tt * tt + 1.0f);
                float sn  = tt * cs;
                er = rot ? er : 1.0f;  ei = rot ? ei : 0.0f;
                cs = rot ? cs : 1.0f;  sn = rot ? sn : 0.0f;
                float ar = er * cs, ai = ei * cs;         // alpha = e^{i phi} cs
                float br = er * sn, bi = ei * sn;         // beta  = e^{i phi} sn
                #pragma unroll
                for (int m = 0; m < 4; ++m) {
                    float pr = wr[m][p], pi = wi[m][p];
                    float qr = wr[m][q], qi = wi[m][q];
                    wr[m][p] =  ar * pr - ai * pi + sn * qr;   // w_p' = a*w_p + sn*w_q
                    wi[m][p] =  ar * pi + ai * pr + sn * qi;
                    wr[m][q] = -(br * pr - bi * pi) + cs * qr; // w_q' = -b*w_p + cs*w_q
                    wi[m][q] = -(br * pi + bi * pr) + cs * qi;
                }
            }
        }
    }

    // ---- publish W and lambda to LDS for the WMMA projection phase ----
    #pragma unroll
    for (int k = 0; k < 4; ++k) {
        float lam = 0.f;
        #pragma unroll
        for (int m = 0; m < 4; ++m) lam += wr[m][k] * wr[m][k] + wi[m][k] * wi[m][k];
        lds_lam[t * 4 + k] = lam;
    }
    #pragma unroll
    for (int m = 0; m < 4; ++m)
        #pragma unroll
        for (int k = 0; k < 4; ++k) {
            lds_w[t * WSTRIDE + (0 * 4 + m) * 4 + k] = wr[m][k];
            lds_w[t * WSTRIDE + (1 * 4 + m) * 4 + k] = wi[m][k];
        }
    __syncthreads();

    // ---- WMMA phase: numerator[k] = conj(W[:,k]) . y, 4 sites per wave-round ----
    // V_WMMA_F32_16X16X4_F32: A 16x4 (M=site*4+k, K=m), B 4x16 (K=m, N=site*4+*),
    // B columns within a site group broadcast y so result is column-invariant.
    const int lane = t & 31;
    const int wv   = t >> 5;
    const int half = lane >> 4;        // A/B K split: K={0,1} lanes 0-15, K={2,3} lanes 16-31
    const int l16  = lane & 15;
    const int sq   = l16 >> 2;         // quad-site 0..3 within a round
    const int kA   = lane & 3;         // A column index (row M = l16 = sq*4+kA)
    const int m0   = half * 2;

    for (int r = 0; r < 8; ++r) {
        const int sA = wv * 32 + r * 4 + sq;   // site-in-block this lane feeds
        v2f a_r, a_i, b_r, b_i;
        a_r[0] = lds_w[sA * WSTRIDE + ((m0 + 0)     ) * 4 + kA];
        a_r[1] = lds_w[sA * WSTRIDE + ((m0 + 1)     ) * 4 + kA];
        a_i[0] = lds_w[sA * WSTRIDE + ((m0 + 0) + 4 ) * 4 + kA];
        a_i[1] = lds_w[sA * WSTRIDE + ((m0 + 1) + 4 ) * 4 + kA];
        b_r[0] = lds_y[((m0 + 0) * 2 + 0) * YSTRIDE + sA];
        b_r[1] = lds_y[((m0 + 1) * 2 + 0) * YSTRIDE + sA];
        b_i[0] = lds_y[((m0 + 0) * 2 + 1) * YSTRIDE + sA];
        b_i[1] = lds_y[((m0 + 1) * 2 + 1) * YSTRIDE + sA];

        v8f cre = {}; v8f cim = {};
        // Re(conj(W).y) = Wr.yr + Wi.yi
        cre = __builtin_amdgcn_wmma_f32_16x16x4_f32(false, a_r, false, b_r,
                                                    (short)0, cre, false, false);
        cre = __builtin_amdgcn_wmma_f32_16x16x4_f32(false, a_i, false, b_i,
                                                    (short)0, cre, false, false);
        // Im(conj(W).y) = Wr.yi - Wi.yr   (f32 WMMA has no A-neg: negate in VALU)
        cim = __builtin_amdgcn_wmma_f32_16x16x4_f32(false, a_r, false, b_i,
                                                    (short)0, cim, false, false);
        v2f a_in; a_in[0] = -a_i[0]; a_in[1] = -a_i[1];
        cim = __builtin_amdgcn_wmma_f32_16x16x4_f32(false, a_in, false, b_r,
                                                    (short)0, cim, false, false);

        // ---- extraction: one lane per site holds the full 4-vector in static VGPRs
        // sites 0,1 -> rows M=0..7 (lanes<16); sites 2,3 -> rows M=8..15 (lanes>=16)
        if (kA == 0 && half == (sq >> 1)) {
            const int sIB = wv * 32 + r * 4 + sq;
            float lam[4], xr[4], xi[4];
            #pragma unroll
            for (int k = 0; k < 4; ++k) {
                xr[k]  = (sq & 1) ? cre[4 + k] : cre[k];
                xi[k]  = (sq & 1) ? cim[4 + k] : cim[k];
                lam[k] = lds_lam[sIB * 4 + k];
            }
            // sort singular triplets descending by lambda = s^2 (numpy SVD order)
            #define CSWP(i, j) { bool sw_ = lam[i] < lam[j];                          \
                float tl = lam[i], tr = xr[i], ti = xi[i];                            \
                lam[i] = sw_ ? lam[j] : lam[i]; xr[i] = sw_ ? xr[j] : xr[i];          \
                xi[i]  = sw_ ? xi[j]  : xi[i];                                        \
                lam[j] = sw_ ? tl : lam[j]; xr[j] = sw_ ? tr : xr[j];                 \
                xi[j]  = sw_ ? ti : xi[j]; }
            CSWP(0, 1) CSWP(2, 3) CSWP(0, 2) CSWP(1, 3) CSWP(1, 2)
            #undef CSWP
            #pragma unroll
            for (int k = 0; k < 4; ++k) {
                float il = __builtin_amdgcn_rcpf(lam[k]);
                int cidx = ((b * 4 + k) * S_ + s) * F_ + f0 + sIB;
                out2[cidx] = make_float2(xr[k] * il, xi[k] * il);
            }
        }
    }
}

extern "C" void kernel_launch(void* const* d_in, const int* in_sizes, int n_in,
                              void* d_out, int out_size, void* d_ws, size_t ws_size,
                              hipStream_t stream) {
    (void)in_sizes; (void)n_in; (void)d_ws; (void)ws_size; (void)out_size;
    const float* yr = (const float*)d_in[0];
    const float* yi = (const float*)d_in[1];
    const float* hr = (const float*)d_in[2];
    const float* hi = (const float*)d_in[3];
    // d_in[4] = precoding_ind: all zeros with NRX==1 -> identity gather, folded in.
    float2* out = (float2*)d_out;   // complex64 = interleaved float pairs

    const int blocks = B_ * S_ * (F_ / 256);   // 16*14*16 = 3584
    svd_eq_jacobi_wmma_kernel<<<blocks, 256, 0, stream>>>(yr, yi, hr, hi, out);
}